// NeuralLDPCDecoder_76330158785035
// MI455X (gfx1250) — compile-verified
//
#include <hip/hip_runtime.h>
#include <stdint.h>

// ---- problem constants (match reference) ----
#define N_V   8192
#define N_C   4096
#define DV    3
#define DC    6
#define N_E   (N_V * DV)      // 24576
#define BATCH 1024
#define ITERS 10
#define CLIPV 0.999f
#define LN2F  0.69314718055994530942f

typedef float v4f __attribute__((ext_vector_type(4)));

// gfx1250 native tanh (TRANS unit) -- new CDNA5 instruction
__device__ __forceinline__ float dev_tanh(float x) {
  float r;
  asm("v_tanh_f32 %0, %1" : "=v"(r) : "v"(x));
  return r;
}
// v_log_f32 (base-2)
__device__ __forceinline__ float dev_log2(float x) { return __builtin_amdgcn_logf(x); }
__device__ __forceinline__ float clipf(float x) { return fminf(fmaxf(x, -CLIPV), CLIPV); }

// -------------------------------------------------------------------------
// Adjacency build: check -> 6 edge ids.  Atomic slot grab, then per-check
// ascending sort => deterministic reduction order independent of atomics.
// -------------------------------------------------------------------------
__global__ void fill_adj_kernel(const int* __restrict__ edge_check,
                                int* __restrict__ chk_edges,
                                int* __restrict__ cnt) {
  int e = blockIdx.x * blockDim.x + threadIdx.x;
  if (e < N_E) {
    int c = edge_check[e];
    int slot = atomicAdd(&cnt[c], 1);
    chk_edges[c * DC + slot] = e;
  }
}

__global__ void sort_adj_kernel(int* __restrict__ chk_edges) {
  int c = blockIdx.x * blockDim.x + threadIdx.x;
  if (c >= N_C) return;
  int v[DC];
#pragma unroll
  for (int j = 0; j < DC; ++j) v[j] = chk_edges[c * DC + j];
  // bubble network, 15 compare-exchanges, fully unrolled
#pragma unroll
  for (int i = 0; i < DC - 1; ++i)
#pragma unroll
    for (int j = 0; j < DC - 1 - i; ++j) {
      int lo = min(v[j], v[j + 1]);
      int hi = max(v[j], v[j + 1]);
      v[j] = lo; v[j + 1] = hi;
    }
#pragma unroll
  for (int j = 0; j < DC; ++j) chk_edges[c * DC + j] = v[j];
}

// -------------------------------------------------------------------------
// llr_t[N][B] = 2 * received[B][N]^T  (also seeds tot = llr_t).
// Tiles staged to LDS with the CDNA5 async-to-LDS path.
// block (32,8), grid (N/32, B/32)
// -------------------------------------------------------------------------
__global__ __launch_bounds__(256) void init_llrt_kernel(const float* __restrict__ received,
                                                        float* __restrict__ llr_t,
                                                        float* __restrict__ tot) {
  __shared__ float tile[32][33];
  const int vt = blockIdx.x * 32;
  const int bt = blockIdx.y * 32;
  const int x = threadIdx.x, y = threadIdx.y;

  // each thread async-loads one 16B chunk of a tile row (row = b_local)
  const int row = y * 4 + (x >> 3);   // 0..31
  const int chunk = x & 7;            // 0..7 (16B chunks)
  const uint64_t ga = (uint64_t)(received + (size_t)(bt + row) * N_V + vt + chunk * 4);
  const unsigned lds_off = (unsigned)(uintptr_t)(&tile[row][chunk * 4]);
  asm volatile("global_load_async_to_lds_b128 %0, %1, off"
               :: "v"(lds_off), "v"(ga) : "memory");
  asm volatile("s_wait_asynccnt 0" ::: "memory");
  __syncthreads();

#pragma unroll
  for (int r = 0; r < 4; ++r) {
    const int vl = y * 4 + r;                 // v_local
    const float val = 2.0f * tile[x][vl];     // tile[b_local][v_local]
    const size_t o = (size_t)(vt + vl) * BATCH + bt + x;
    llr_t[o] = val;
    tot[o]   = val;
  }
}

// -------------------------------------------------------------------------
// Check-node update (in-place on c2v): one block per check, full batch.
// v2c[e] = tot[v] - c2v_old[e];  t = tanh(v2c/2);
// loo via prefix/suffix products (no divides);
// c2v_new = ln2 * (log2(1+clip(loo)) - log2(1-clip(loo)))  == 2*atanh(clip)
// -------------------------------------------------------------------------
__global__ __launch_bounds__(256) void check_kernel(const float* __restrict__ tot,
                                                    float* __restrict__ c2v,
                                                    const int* __restrict__ chk_edges) {
  const int bo = threadIdx.x * 4;   // batch offset (float4)
  const int c = blockIdx.x;

  int e[DC];
#pragma unroll
  for (int j = 0; j < DC; ++j) e[j] = chk_edges[c * DC + j];   // uniform -> scalar loads

  v4f t[DC], pre[DC];
  v4f run = {1.f, 1.f, 1.f, 1.f};
#pragma unroll
  for (int j = 0; j < DC; ++j) {
    const v4f tv = *(const v4f*)(tot + (size_t)(e[j] / DV) * BATCH + bo);
    const v4f cv = *(const v4f*)(c2v + (size_t)e[j] * BATCH + bo);
    const v4f m = 0.5f * (tv - cv);
    v4f tj;
#pragma unroll
    for (int k = 0; k < 4; ++k) tj[k] = dev_tanh(m[k]);
    t[j] = tj;
    pre[j] = run;
    run *= tj;
  }
  v4f suf = {1.f, 1.f, 1.f, 1.f};
#pragma unroll
  for (int j = DC - 1; j >= 0; --j) {
    const v4f loo = pre[j] * suf;
    suf *= t[j];
    v4f o;
#pragma unroll
    for (int k = 0; k < 4; ++k) {
      const float xx = clipf(loo[k]);
      o[k] = LN2F * (dev_log2(1.0f + xx) - dev_log2(1.0f - xx));
    }
    *(v4f*)(c2v + (size_t)e[j] * BATCH + bo) = o;
  }
}

// -------------------------------------------------------------------------
// Variable-node totals: tot[v] = (c2v[3v]+c2v[3v+1]+c2v[3v+2]) + llr_t[v]
// one block per variable, pure streaming.
// -------------------------------------------------------------------------
__global__ __launch_bounds__(256) void tot_kernel(const float* __restrict__ llr_t,
                                                  const float* __restrict__ c2v,
                                                  float* __restrict__ tot) {
  const int bo = threadIdx.x * 4;
  const int v = blockIdx.x;
  const v4f a = *(const v4f*)(c2v + (size_t)(3 * v + 0) * BATCH + bo);
  const v4f b = *(const v4f*)(c2v + (size_t)(3 * v + 1) * BATCH + bo);
  const v4f cc = *(const v4f*)(c2v + (size_t)(3 * v + 2) * BATCH + bo);
  const v4f l = *(const v4f*)(llr_t + (size_t)v * BATCH + bo);
  const v4f s = (a + b + cc) + l;
  *(v4f*)(tot + (size_t)v * BATCH + bo) = s;
}

// -------------------------------------------------------------------------
// Final posteriors + bits, transposed [b][v] via padded LDS tile.
// block (32,8), grid (N/32, B/32)
// -------------------------------------------------------------------------
__global__ __launch_bounds__(256) void final_kernel(const float* __restrict__ llr_t,
                                                    const float* __restrict__ c2v,
                                                    float* __restrict__ out_llr,
                                                    int* __restrict__ out_bits) {
  __shared__ float tile[32][33];
  const int vt = blockIdx.x * 32;
  const int bt = blockIdx.y * 32;
  const int x = threadIdx.x, y = threadIdx.y;

#pragma unroll
  for (int r = 0; r < 4; ++r) {
    const int vl = y * 4 + r;
    const int v = vt + vl;
    const size_t col = (size_t)bt + x;
    const float s = (c2v[(size_t)(3 * v + 0) * BATCH + col] +
                     c2v[(size_t)(3 * v + 1) * BATCH + col] +
                     c2v[(size_t)(3 * v + 2) * BATCH + col]) +
                    llr_t[(size_t)v * BATCH + col];
    tile[x][vl] = s;   // tile[b_local][v_local]
  }
  __syncthreads();
#pragma unroll
  for (int r = 0; r < 4; ++r) {
    const int bl = y * 4 + r;
    const float s = tile[bl][x];
    const size_t o = (size_t)(bt + bl) * N_V + vt + x;
    out_llr[o] = s;
    out_bits[o] = (s < 0.0f) ? 1 : 0;
  }
}

// -------------------------------------------------------------------------
extern "C" void kernel_launch(void* const* d_in, const int* in_sizes, int n_in,
                              void* d_out, int out_size, void* d_ws, size_t ws_size,
                              hipStream_t stream) {
  (void)in_sizes; (void)n_in; (void)out_size; (void)ws_size;

  const float* received = (const float*)d_in[0];
  // d_in[1] = edge_var (implicit: var = e / 3), unused
  const int* edge_check = (const int*)d_in[2];

  char* ws = (char*)d_ws;
  float* c2v   = (float*)ws;                                         // E*B floats (100.7 MB)
  float* llr_t = (float*)(ws + (size_t)N_E * BATCH * 4);             // N*B floats (33.5 MB)
  float* tot   = (float*)(ws + (size_t)(N_E + N_V) * BATCH * 4);     // N*B floats (33.5 MB)
  int* chk_edges = (int*)(ws + (size_t)(N_E + 2 * N_V) * BATCH * 4); // R*6 ints
  int* cnt = chk_edges + (size_t)N_C * DC;                           // R ints

  hipMemsetAsync(c2v, 0, (size_t)N_E * BATCH * sizeof(float), stream);
  hipMemsetAsync(cnt, 0, (size_t)N_C * sizeof(int), stream);

  fill_adj_kernel<<<(N_E + 255) / 256, 256, 0, stream>>>(edge_check, chk_edges, cnt);
  sort_adj_kernel<<<(N_C + 255) / 256, 256, 0, stream>>>(chk_edges);
  init_llrt_kernel<<<dim3(N_V / 32, BATCH / 32), dim3(32, 8), 0, stream>>>(received, llr_t, tot);

  for (int it = 0; it < ITERS; ++it) {
    check_kernel<<<N_C, 256, 0, stream>>>(tot, c2v, chk_edges);
    if (it + 1 < ITERS)   // last variable update is folded into final_kernel
      tot_kernel<<<N_V, 256, 0, stream>>>(llr_t, c2v, tot);
  }

  float* out_llr = (float*)d_out;
  int* out_bits = (int*)((float*)d_out + (size_t)BATCH * N_V);
  final_kernel<<<dim3(N_V / 32, BATCH / 32), dim3(32, 8), 0, stream>>>(llr_t, c2v, out_llr, out_bits);
}